// ResidualVectorQuantizer_10788957847718
// MI455X (gfx1250) — compile-verified
//
#include <hip/hip_runtime.h>

typedef __attribute__((ext_vector_type(2))) float v2f;
typedef __attribute__((ext_vector_type(8))) float v8f;

#define NTOK   65536          // 16*4096 tokens
#define DIMS   512
#define KCODES 1024
#define LAYERS 4
#define TOKPB  128            // tokens per block (8 waves * 16)
#define RS     516            // padded LDS row stride (floats)
#define QN     33554432       // 16*4096*512
#define IDXN   262144         // 4*65536
#define SMEMB  ((TOKPB*RS + KCODES) * 4)

// ---------------- init: zero the loss accumulator ----------------
__global__ void rvq_init(float* S) { *S = 0.0f; }

// ---------------- codebook squared norms: one wave per row ----------------
__global__ void __launch_bounds__(256) rvq_cnorm(const float* __restrict__ cbs,
                                                 float* __restrict__ cn) {
  int row  = blockIdx.x * 8 + (threadIdx.x >> 5);   // 0..4095 (= l*1024 + k)
  int lane = threadIdx.x & 31;
  if (row >= LAYERS * KCODES) return;
  const float4* p = reinterpret_cast<const float4*>(cbs + (long)row * DIMS);
  float s = 0.0f;
#pragma unroll
  for (int it = 0; it < 4; ++it) {
    float4 v = p[lane + it * 32];
    s += v.x * v.x + v.y * v.y + v.z * v.z + v.w * v.w;
  }
  s += __shfl_xor(s, 16, 32); s += __shfl_xor(s, 8, 32);
  s += __shfl_xor(s, 4, 32);  s += __shfl_xor(s, 2, 32);
  s += __shfl_xor(s, 1, 32);
  if (lane == 0) cn[row] = s;
}

// ---------------- main fused RVQ kernel ----------------
__global__ void __launch_bounds__(256) rvq_main(const float* __restrict__ x,
                                                const float* __restrict__ cbs,
                                                const float* __restrict__ cnorm,
                                                float* __restrict__ out,
                                                float* __restrict__ Sacc) {
  extern __shared__ float smem[];
  float* Rsh = smem;                 // [TOKPB][RS] residual, fp32
  float* cns = smem + TOKPB * RS;    // [KCODES] this layer's |c|^2

  const int tid   = threadIdx.x;
  const int lane  = tid & 31;
  const int w     = tid >> 5;        // wave id 0..7
  const int l16   = lane & 15;
  const int halo  = lane >> 4;       // 0 or 1
  const long tok0 = (long)blockIdx.x * TOKPB;
  const int  myrow0 = w * 16;
  float* myR = &Rsh[(myrow0 + l16) * RS];

  // cooperative load x -> residual in LDS (float4 lanes)
  for (int i = tid; i < TOKPB * (DIMS / 4); i += 256) {
    int row = i >> 7, c4 = i & 127;
    float4 v = reinterpret_cast<const float4*>(x + (tok0 + row) * DIMS)[c4];
    *reinterpret_cast<float4*>(&Rsh[row * RS + c4 * 4]) = v;
  }

  float lsum = 0.0f;

  for (int l = 0; l < LAYERS; ++l) {
    __syncthreads();                                   // R ready / cns free
    for (int i = tid; i < KCODES; i += 256) cns[i] = cnorm[l * KCODES + i];
    __syncthreads();
    const float* cbL = cbs + (long)l * KCODES * DIMS;

    float minv = 3.4e38f;
    int   mini = 0;

    for (int kt = 0; kt < KCODES; kt += 64) {          // four 16-code tiles
      const float* arow = cbL + (long)(kt + l16) * DIMS;
      v8f acc0 = {}; v8f acc1 = {}; v8f acc2 = {}; v8f acc3 = {};
#pragma unroll 2
      for (int c8 = 0; c8 < DIMS / 8; ++c8) {
        const int db = c8 * 8 + (halo << 2);           // permuted-K b128 trick
        float4 a0 = *reinterpret_cast<const float4*>(arow + db);
        float4 a1 = *reinterpret_cast<const float4*>(arow + 16 * DIMS + db);
        float4 a2 = *reinterpret_cast<const float4*>(arow + 32 * DIMS + db);
        float4 a3 = *reinterpret_cast<const float4*>(arow + 48 * DIMS + db);
        float4 b  = *reinterpret_cast<const float4*>(myR + db);
        v2f fblo = {b.x, b.y}, fbhi = {b.z, b.w};
        v2f t0lo = {a0.x, a0.y}, t0hi = {a0.z, a0.w};
        v2f t1lo = {a1.x, a1.y}, t1hi = {a1.z, a1.w};
        v2f t2lo = {a2.x, a2.y}, t2hi = {a2.z, a2.w};
        v2f t3lo = {a3.x, a3.y}, t3hi = {a3.z, a3.w};
        acc0 = __builtin_amdgcn_wmma_f32_16x16x4_f32(false, t0lo, false, fblo, (short)0, acc0, false, false);
        acc1 = __builtin_amdgcn_wmma_f32_16x16x4_f32(false, t1lo, false, fblo, (short)0, acc1, false, false);
        acc2 = __builtin_amdgcn_wmma_f32_16x16x4_f32(false, t2lo, false, fblo, (short)0, acc2, false, false);
        acc3 = __builtin_amdgcn_wmma_f32_16x16x4_f32(false, t3lo, false, fblo, (short)0, acc3, false, false);
        acc0 = __builtin_amdgcn_wmma_f32_16x16x4_f32(false, t0hi, false, fbhi, (short)0, acc0, false, false);
        acc1 = __builtin_amdgcn_wmma_f32_16x16x4_f32(false, t1hi, false, fbhi, (short)0, acc1, false, false);
        acc2 = __builtin_amdgcn_wmma_f32_16x16x4_f32(false, t2hi, false, fbhi, (short)0, acc2, false, false);
        acc3 = __builtin_amdgcn_wmma_f32_16x16x4_f32(false, t3hi, false, fbhi, (short)0, acc3, false, false);
      }
      // lane holds C[code][token]; codes ascend with j and tile -> first-min tie rule holds
      const int cb0 = kt + (halo << 3);
#pragma unroll
      for (int j = 0; j < 8; ++j) {
        float s0 = cns[cb0 + j] - 2.0f * acc0[j];
        if (s0 < minv) { minv = s0; mini = cb0 + j; }
      }
#pragma unroll
      for (int j = 0; j < 8; ++j) {
        float s1 = cns[cb0 + 16 + j] - 2.0f * acc1[j];
        if (s1 < minv) { minv = s1; mini = cb0 + 16 + j; }
      }
#pragma unroll
      for (int j = 0; j < 8; ++j) {
        float s2 = cns[cb0 + 32 + j] - 2.0f * acc2[j];
        if (s2 < minv) { minv = s2; mini = cb0 + 32 + j; }
      }
#pragma unroll
      for (int j = 0; j < 8; ++j) {
        float s3 = cns[cb0 + 48 + j] - 2.0f * acc3[j];
        if (s3 < minv) { minv = s3; mini = cb0 + 48 + j; }
      }
    }

    // merge the two lane-halves (codes split 0-7 / 8-15 per tile)
    {
      float ov = __shfl_xor(minv, 16, 32);
      int   oi = __shfl_xor(mini, 16, 32);
      if (ov < minv || (ov == minv && oi < mini)) { minv = ov; mini = oi; }
    }

    if (lane < 16)
      out[QN + (long)l * NTOK + tok0 + myrow0 + l16] = (float)mini;

    // gather quant row, accumulate loss, update residual (own rows only)
    for (int t = 0; t < 16; ++t) {
      const int qi = __shfl(mini, t, 32);
      const float* qrow = cbL + (long)qi * DIMS;
      float* rrow = &Rsh[(myrow0 + t) * RS];
#pragma unroll
      for (int it = 0; it < 4; ++it) {
        int d = (lane + it * 32) * 4;
        float4 q = *reinterpret_cast<const float4*>(qrow + d);
        float4 r = *reinterpret_cast<float4*>(rrow + d);
        float dx = q.x - r.x, dy = q.y - r.y, dz = q.z - r.z, dw = q.w - r.w;
        lsum += dx * dx + dy * dy + dz * dz + dw * dw;
        r.x -= q.x; r.y -= q.y; r.z -= q.z; r.w -= q.w;
        *reinterpret_cast<float4*>(rrow + d) = r;
      }
    }
  }

  __syncthreads();
  // quantized = x - residual_final (cooperative, float4)
  for (int i = tid; i < TOKPB * (DIMS / 4); i += 256) {
    int row = i >> 7, c4 = i & 127;
    float4 xv = reinterpret_cast<const float4*>(x + (tok0 + row) * DIMS)[c4];
    float4 rv = *reinterpret_cast<float4*>(&Rsh[row * RS + c4 * 4]);
    float4 qv = make_float4(xv.x - rv.x, xv.y - rv.y, xv.z - rv.z, xv.w - rv.w);
    reinterpret_cast<float4*>(out + (tok0 + row) * DIMS)[c4] = qv;
  }

  // loss reduction
  lsum += __shfl_xor(lsum, 16, 32); lsum += __shfl_xor(lsum, 8, 32);
  lsum += __shfl_xor(lsum, 4, 32);  lsum += __shfl_xor(lsum, 2, 32);
  lsum += __shfl_xor(lsum, 1, 32);
  if (lane == 0) atomicAdd(Sacc, lsum);
}

// ---------------- finalize scalars ----------------
__global__ void rvq_fin(const float* S, float* out) {
  float s   = *S;
  float cbl = s * (1.0f / (float)QN);   // sum of per-layer means (numel = QN)
  float cmt = 0.25f * cbl;
  out[QN + IDXN + 0] = cmt;
  out[QN + IDXN + 1] = cbl;
  out[QN + IDXN + 2] = cmt + cbl;
}

extern "C" void kernel_launch(void* const* d_in, const int* in_sizes, int n_in,
                              void* d_out, int out_size, void* d_ws, size_t ws_size,
                              hipStream_t stream) {
  const float* x   = (const float*)d_in[0];   // [16,4096,512]
  const float* cbs = (const float*)d_in[1];   // [4,1024,512]
  float* out = (float*)d_out;
  float* wsf = (float*)d_ws;
  float* S   = wsf;                            // 1 float
  float* cn  = wsf + 16;                       // 4096 floats

  rvq_init<<<1, 1, 0, stream>>>(S);
  rvq_cnorm<<<(LAYERS * KCODES) / 8, 256, 0, stream>>>(cbs, cn);

  hipFuncSetAttribute(reinterpret_cast<const void*>(rvq_main),
                      hipFuncAttributeMaxDynamicSharedMemorySize, SMEMB);
  rvq_main<<<NTOK / TOKPB, 256, SMEMB, stream>>>(x, cbs, cn, out, S);

  rvq_fin<<<1, 1, 0, stream>>>(S, out);
}